// Net_5789615915291
// MI455X (gfx1250) — compile-verified
//
#include <hip/hip_runtime.h>
#include <math.h>

typedef __attribute__((ext_vector_type(2))) float v2f;
typedef __attribute__((ext_vector_type(8))) float v8f;

// ---------------------------------------------------------------- threefry
__device__ __forceinline__ unsigned rotl32(unsigned x, int r) {
    return (x << r) | (x >> (32 - r));
}

// JAX threefry2x32 with key = PRNGKey(42) = {0, 42}
__device__ __forceinline__ void threefry42(unsigned& x0, unsigned& x1) {
    const unsigned k0 = 0u, k1 = 42u;
    const unsigned k2 = 0x1BD11BDAu ^ k0 ^ k1;
    x0 += k0; x1 += k1;
#define TF_R4(a,b,c,d)  \
    x0 += x1; x1 = rotl32(x1,a); x1 ^= x0; \
    x0 += x1; x1 = rotl32(x1,b); x1 ^= x0; \
    x0 += x1; x1 = rotl32(x1,c); x1 ^= x0; \
    x0 += x1; x1 = rotl32(x1,d); x1 ^= x0;
    TF_R4(13,15,26,6);  x0 += k1; x1 += k2 + 1u;
    TF_R4(17,29,16,24); x0 += k2; x1 += k0 + 2u;
    TF_R4(13,15,26,6);  x0 += k0; x1 += k1 + 3u;
    TF_R4(17,29,16,24); x0 += k1; x1 += k2 + 4u;
    TF_R4(13,15,26,6);  x0 += k2; x1 += k0 + 5u;
#undef TF_R4
}

__device__ __forceinline__ float drop_apply(float v, unsigned bits) {
    // JAX uniform: bitcast((bits>>9)|0x3f800000) - 1.0 ; keep if < 0.8
    float u = __uint_as_float((bits >> 9) | 0x3f800000u) - 1.0f;
    return (u < 0.8f) ? v * 1.25f : 0.0f;
}

// ---------------------------------------------------------------- degrees
__global__ __launch_bounds__(256) void degree_kernel(const long long* __restrict__ ei,
                                                     float* __restrict__ deg_src,
                                                     float* __restrict__ deg_dst, int E) {
    int e = blockIdx.x * 256 + threadIdx.x;
    if (e < E) {
        atomicAdd(&deg_src[(unsigned)(int)ei[e]], 1.0f);
        atomicAdd(&deg_dst[(unsigned)(int)ei[E + e]], 1.0f);
    }
}

__global__ __launch_bounds__(256) void norm_kernel(float* __restrict__ deg, int n2) {
    int i = blockIdx.x * 256 + threadIdx.x;
    if (i < n2) deg[i] = rsqrtf(fmaxf(deg[i], 1.0f));
}

// ---------------------------------------------------------------- GEMM1: h = (x*ns) @ W1  [256 -> 128], 32-row M-tile
#define LDA1 260   // stride 260 (bank-conflict free: 260 % 64 == 4; 1040 B rows, 16B aligned)
__global__ __launch_bounds__(256) void gemm1_kernel(const float* __restrict__ x,
                                                    const float* __restrict__ norm_src,
                                                    const float* __restrict__ W1,
                                                    float* __restrict__ h, int n) {
    __shared__ float As[32 * LDA1];
    const int tid = threadIdx.x;
    const int rb  = blockIdx.x * 32;
    // stage 32x256 tile of x, pre-scaled by norm_src (each thread: 8 x float4)
    {
        const int row  = tid >> 3;           // 0..31
        const int cb   = (tid & 7) * 32;     // 0..224
        const int grow = rb + row;
        float ns = 0.0f;
        float4 z = make_float4(0.f, 0.f, 0.f, 0.f);
        float4 v[8] = {z, z, z, z, z, z, z, z};
        if (grow < n) {
            ns = norm_src[grow];
            const float4* xr = (const float4*)(x + (size_t)grow * 256 + cb);
#pragma unroll
            for (int q = 0; q < 8; q++) v[q] = xr[q];
        }
        float* dl = &As[row * LDA1 + cb];
#pragma unroll
        for (int q = 0; q < 8; q++) {
            dl[q * 4 + 0] = v[q].x * ns;
            dl[q * 4 + 1] = v[q].y * ns;
            dl[q * 4 + 2] = v[q].z * ns;
            dl[q * 4 + 3] = v[q].w * ns;
        }
    }
    __syncthreads();

    const int lane = tid & 31;
    const int wave = tid >> 5;           // 0..7 -> N-tile
    const int m    = lane & 15;
    const int koff = (lane >> 4) * 2;    // 0 or 2 (ISA 16x4 f32 A layout)
    const int col  = wave * 16 + m;
    const float* Ar0 = &As[m * LDA1];
    const float* Ar1 = &As[(m + 16) * LDA1];

    v8f acc0 = {}, acc1 = {};
#pragma unroll 4
    for (int k0 = 0; k0 < 256; k0 += 4) {
        v2f b, a0, a1;
        b.x  = W1[(k0 + koff) * 128 + col];
        b.y  = W1[(k0 + koff + 1) * 128 + col];
        a0.x = Ar0[k0 + koff];
        a0.y = Ar0[k0 + koff + 1];
        a1.x = Ar1[k0 + koff];
        a1.y = Ar1[k0 + koff + 1];
        acc0 = __builtin_amdgcn_wmma_f32_16x16x4_f32(false, a0, false, b,
                                                     (short)0, acc0, false, false);
        acc1 = __builtin_amdgcn_wmma_f32_16x16x4_f32(false, a1, false, b,
                                                     (short)0, acc1, false, false);
    }
    const int r0 = rb + (lane >> 4) * 8;                 // rows r0..r0+7 and r0+16..r0+23
    if (rb + 32 <= n) {                                  // fast path: full tile (no guards)
#pragma unroll
        for (int v = 0; v < 8; v++) h[(r0 + v) * 128 + col]      = acc0[v];
#pragma unroll
        for (int v = 0; v < 8; v++) h[(r0 + 16 + v) * 128 + col] = acc1[v];
    } else {
#pragma unroll
        for (int v = 0; v < 8; v++) if (r0 + v < n)      h[(r0 + v) * 128 + col]      = acc0[v];
#pragma unroll
        for (int v = 0; v < 8; v++) if (r0 + 16 + v < n) h[(r0 + 16 + v) * 128 + col] = acc1[v];
    }
}

// ---------------------------------------------------------------- scatter1: agg1[dst] += h[src]  (128 f32/edge)
__global__ __launch_bounds__(256) void scatter1_kernel(const long long* __restrict__ ei,
                                                       const float* __restrict__ h,
                                                       float* __restrict__ agg, int E) {
    const int wave = threadIdx.x >> 5;
    const int lane = threadIdx.x & 31;
    const int e = blockIdx.x * 8 + wave;
    if (e >= E) return;
    const unsigned s = (unsigned)(int)ei[e];
    const unsigned d = (unsigned)(int)ei[E + e];
    const float4 v = *(const float4*)(h + s * 128u + (unsigned)(lane * 4));
    float* o = agg + d * 128u + (unsigned)(lane * 4);
    atomicAdd(o + 0, v.x);
    atomicAdd(o + 1, v.y);
    atomicAdd(o + 2, v.z);
    atomicAdd(o + 3, v.w);
}

// ---------------------------------------------------------------- post1: norm+bias, relu, exact JAX dropout (in place)
__global__ __launch_bounds__(256) void post1_kernel(float* __restrict__ a,
                                                    const float* __restrict__ norm_dst,
                                                    const float* __restrict__ b1, int n) {
    const unsigned half = (unsigned)n * 64u;      // (n*128)/2 counter pairs
    unsigned t = blockIdx.x * 256 + threadIdx.x;
    if (t >= half) return;
    unsigned x0 = t, x1 = t + half;
    threefry42(x0, x1);
    {
        unsigned idx = t;
        float v = fmaxf(a[idx] * norm_dst[idx >> 7] + b1[idx & 127u], 0.0f);
        a[idx] = drop_apply(v, x0);
    }
    {
        unsigned idx = t + half;
        float v = fmaxf(a[idx] * norm_dst[idx >> 7] + b1[idx & 127u], 0.0f);
        a[idx] = drop_apply(v, x1);
    }
}

// ---------------------------------------------------------------- GEMM2: h2 = (h1*ns) @ W2  [128 -> 47, pad 48]
#define LDA2 132
__global__ __launch_bounds__(128) void gemm2_kernel(const float* __restrict__ h1,
                                                    const float* __restrict__ norm_src,
                                                    const float* __restrict__ W2,
                                                    float* __restrict__ h2, int n) {
    __shared__ float As[16 * LDA2];
    const int tid = threadIdx.x;
    const int rb  = blockIdx.x * 16;
    {
        const int row  = tid >> 3;           // 0..15
        const int cb   = (tid & 7) * 16;     // 0..112
        const int grow = rb + row;
        float ns = 0.0f;
        float4 z = make_float4(0.f, 0.f, 0.f, 0.f);
        float4 v0 = z, v1 = z, v2 = z, v3 = z;
        if (grow < n) {
            ns = norm_src[grow];
            const float4* hr = (const float4*)(h1 + (size_t)grow * 128 + cb);
            v0 = hr[0]; v1 = hr[1]; v2 = hr[2]; v3 = hr[3];
        }
        float* dl = &As[row * LDA2 + cb];
        dl[0]  = v0.x*ns; dl[1]  = v0.y*ns; dl[2]  = v0.z*ns; dl[3]  = v0.w*ns;
        dl[4]  = v1.x*ns; dl[5]  = v1.y*ns; dl[6]  = v1.z*ns; dl[7]  = v1.w*ns;
        dl[8]  = v2.x*ns; dl[9]  = v2.y*ns; dl[10] = v2.z*ns; dl[11] = v2.w*ns;
        dl[12] = v3.x*ns; dl[13] = v3.y*ns; dl[14] = v3.z*ns; dl[15] = v3.w*ns;
    }
    __syncthreads();

    const int lane = tid & 31;
    const int wave = tid >> 5;               // waves 0..2 cover 48 cols
    if (wave >= 3) return;
    const int m    = lane & 15;
    const int koff = (lane >> 4) * 2;
    const int col  = wave * 16 + m;
    const bool cv  = col < 47;
    const float* Ar = &As[m * LDA2];

    v8f acc = {};
#pragma unroll 4
    for (int k0 = 0; k0 < 128; k0 += 4) {
        v2f a, b;
        a.x = Ar[k0 + koff];
        a.y = Ar[k0 + koff + 1];
        b.x = cv ? W2[(k0 + koff) * 47 + col] : 0.0f;
        b.y = cv ? W2[(k0 + koff + 1) * 47 + col] : 0.0f;
        acc = __builtin_amdgcn_wmma_f32_16x16x4_f32(false, a, false, b,
                                                    (short)0, acc, false, false);
    }
    const int r0 = rb + (lane >> 4) * 8;
    if (cv) {
        if (rb + 16 <= n) {                  // fast path: full tile
#pragma unroll
            for (int v = 0; v < 8; v++) h2[(r0 + v) * 48 + col] = acc[v];
        } else {
#pragma unroll
            for (int v = 0; v < 8; v++) if (r0 + v < n) h2[(r0 + v) * 48 + col] = acc[v];
        }
    }
}

// ---------------------------------------------------------------- scatter2: agg2[dst] += h2[src]  (47 f32/edge, stride 48)
__global__ __launch_bounds__(256) void scatter2_kernel(const long long* __restrict__ ei,
                                                       const float* __restrict__ h2,
                                                       float* __restrict__ agg, int E) {
    const int wave = threadIdx.x >> 5;
    const int lane = threadIdx.x & 31;
    const int e = blockIdx.x * 8 + wave;
    if (e >= E) return;
    const unsigned s = (unsigned)(int)ei[e];
    const unsigned d = (unsigned)(int)ei[E + e];
    const float* src = h2 + s * 48u;
    float* dst = agg + d * 48u;
    atomicAdd(dst + lane, src[lane]);
    if (lane < 15) atomicAdd(dst + lane + 32, src[lane + 32]);
}

// ---------------------------------------------------------------- post2: norm+bias + log_softmax(47) -> out
__global__ __launch_bounds__(256) void post2_kernel(const float* __restrict__ agg,
                                                    const float* __restrict__ norm_dst,
                                                    const float* __restrict__ b2,
                                                    float* __restrict__ out, int n) {
    const int wave = threadIdx.x >> 5;
    const int lane = threadIdx.x & 31;
    const int i = blockIdx.x * 8 + wave;
    if (i >= n) return;
    const float nd = norm_dst[i];
    const float* ar = agg + (unsigned)i * 48u;
    float v0 = ar[lane] * nd + b2[lane];                                     // lane 0..31 < 47
    float v1 = (lane < 15) ? (ar[lane + 32] * nd + b2[lane + 32]) : -INFINITY;
    float mx = fmaxf(v0, v1);
#pragma unroll
    for (int o = 16; o > 0; o >>= 1) mx = fmaxf(mx, __shfl_xor(mx, o, 32));
    float s = expf(v0 - mx) + ((lane < 15) ? expf(v1 - mx) : 0.0f);
#pragma unroll
    for (int o = 16; o > 0; o >>= 1) s += __shfl_xor(s, o, 32);
    const float lse = mx + logf(s);
    float* orow = out + (unsigned)i * 47u;
    orow[lane] = v0 - lse;
    if (lane < 15) orow[lane + 32] = v1 - lse;
}

// ---------------------------------------------------------------- launch
extern "C" void kernel_launch(void* const* d_in, const int* in_sizes, int n_in,
                              void* d_out, int out_size, void* d_ws, size_t ws_size,
                              hipStream_t stream) {
    const float*     x   = (const float*)d_in[0];
    const long long* ei  = (const long long*)d_in[1];   // int64 [2, E]
    const float*     W1  = (const float*)d_in[2];
    const float*     b1  = (const float*)d_in[3];
    const float*     W2  = (const float*)d_in[4];
    const float*     b2  = (const float*)d_in[5];
    float*           out = (float*)d_out;

    const int n = in_sizes[0] / 256;
    const int E = in_sizes[1] / 2;

    // workspace layout (floats):
    //   [0,n)        norm_src (deg -> rsqrt in place)
    //   [n,2n)       norm_dst
    //   [2n, +n*128) h   (layer1 GEMM out; reused: h2 = first n*48, agg2 = next n*48)
    //   [.., +n*128) agg1 (scatter target; reused in place as h1 after post1)
    float* ws       = (float*)d_ws;
    float* norm_src = ws;
    float* norm_dst = ws + n;
    float* h        = ws + 2 * (size_t)n;
    float* agg1     = h + (size_t)n * 128;
    float* h2       = h;                       // reuse (h dead after scatter1)
    float* agg2     = h + (size_t)n * 48;

    hipMemsetAsync(norm_src, 0, 2 * (size_t)n * sizeof(float), stream);
    hipMemsetAsync(agg1, 0, (size_t)n * 128 * sizeof(float), stream);

    degree_kernel<<<(E + 255) / 256, 256, 0, stream>>>(ei, norm_src, norm_dst, E);
    norm_kernel<<<(2 * n + 255) / 256, 256, 0, stream>>>(norm_src, 2 * n);

    gemm1_kernel<<<(n + 31) / 32, 256, 0, stream>>>(x, norm_src, W1, h, n);
    scatter1_kernel<<<(E + 7) / 8, 256, 0, stream>>>(ei, h, agg1, E);
    post1_kernel<<<(n * 64 + 255) / 256, 256, 0, stream>>>(agg1, norm_dst, b1, n);

    hipMemsetAsync(agg2, 0, (size_t)n * 48 * sizeof(float), stream);
    gemm2_kernel<<<(n + 15) / 16, 128, 0, stream>>>(agg1, norm_src, W2, h2, n);
    scatter2_kernel<<<(E + 7) / 8, 256, 0, stream>>>(ei, h2, agg2, E);
    post2_kernel<<<(n + 7) / 8, 256, 0, stream>>>(agg2, norm_dst, b2, out, n);
}